// KarplusStrong_5282809774423
// MI455X (gfx1250) — compile-verified
//
#include <hip/hip_runtime.h>

typedef __attribute__((ext_vector_type(2))) float v2f;
typedef __attribute__((ext_vector_type(8))) float v8f;

#define PIANO_SAMPLES 2621440
#define BLOCK_N 512
#define HOP 256
#define N_STEPS 10240
#define N_FRAMES 10242
#define NBINS 257
#define NE (N_FRAMES * 32)   /* 327744 envelope samples */
#define PCOLS 576            /* 514 (real|imag) padded so N-tiles divide by 64    */
#define KPAD 516             /* 514 padded to multiple of 4 for the K=4 WMMA step */
#define SROWS 10256          /* 10241 spectra padded to multiple of 16            */
#define TWO_PI_OVER_512 0.012271846303085129838f

// ---------------------------------------------------------------------------
// Zero-fill (graph-capture-safe alternative to hipMemsetAsync)
// ---------------------------------------------------------------------------
__global__ void ks_zero(float* __restrict__ p, int n) {
  int i = blockIdx.x * blockDim.x + threadIdx.x;
  if (i < n) p[i] = 0.0f;
}

// ---------------------------------------------------------------------------
// X[t][n] = noise[t][n] * linear_upsample8(excitation_env^2)[t*HOP + n]
// ---------------------------------------------------------------------------
__global__ void ks_gen_x(const float* __restrict__ ev, const float* __restrict__ noise,
                         float* __restrict__ X) {
  int idx = blockIdx.x * blockDim.x + threadIdx.x;
  if (idx >= N_STEPS * BLOCK_N) return;
  int t = idx >> 9;
  int n = idx & 511;
  float c = ((float)(t * HOP + n) + 0.5f) * 0.125f - 0.5f;   // align_corners=False
  c = fminf(fmaxf(c, 0.0f), (float)(NE - 1));
  int lo = (int)c;                                           // c >= 0 -> trunc == floor
  int hi = min(lo + 1, NE - 1);
  float w = c - (float)lo;
  float e0 = ev[lo]; e0 *= e0;                               // square BEFORE interp
  float e1 = ev[hi]; e1 *= e1;
  X[idx] = noise[idx] * (e0 * (1.0f - w) + e1 * w);
}

// ---------------------------------------------------------------------------
// Forward basis with etf folded in:  P = X @ Bf  gives  rfft(x) * etf  directly.
// Bf[n][k]      =  cos(2*pi*k*n/512)*er[k] + sin(2*pi*k*n/512)*ei[k]   (real part)
// Bf[n][257+k]  =  cos(...)*ei[k]          - sin(...)*er[k]            (imag part)
// Padding cols (>=514) are exact zeros.
// ---------------------------------------------------------------------------
__global__ void ks_gen_bf(const float* __restrict__ er, const float* __restrict__ ei,
                          float* __restrict__ Bf) {
  int idx = blockIdx.x * blockDim.x + threadIdx.x;
  if (idx >= 512 * PCOLS) return;
  int n  = idx / PCOLS;
  int kp = idx % PCOLS;
  float v = 0.0f;
  if (kp < 2 * NBINS) {
    int k = (kp < NBINS) ? kp : kp - NBINS;
    int m = (k * n) & 511;                                   // exact angle reduction
    float th = (float)m * TWO_PI_OVER_512;
    float cs = cosf(th), sn = sinf(th);
    v = (kp < NBINS) ? (cs * er[k] + sn * ei[k]) : (cs * ei[k] - sn * er[k]);
  }
  Bf[idx] = v;
}

// ---------------------------------------------------------------------------
// Inverse basis:  blocks = S @ Bi  is irfft (default norm) of each spectrum.
// Bi[k][n] = w_k*cos(2*pi*k*n/512)/512,  Bi[257+k][n] = -w_k*sin(...)/512
// w_0 = w_256 = 1, else 2.  Padding rows (514,515) are zeros.
// ---------------------------------------------------------------------------
__global__ void ks_gen_bi(float* __restrict__ Bi) {
  int idx = blockIdx.x * blockDim.x + threadIdx.x;
  if (idx >= KPAD * 512) return;
  int kp = idx / 512;
  int n  = idx % 512;
  float v = 0.0f;
  if (kp < 2 * NBINS) {
    int k = (kp < NBINS) ? kp : kp - NBINS;
    float w = (k == 0 || k == 256) ? 1.0f : 2.0f;
    int m = (k * n) & 511;
    float th = (float)m * TWO_PI_OVER_512;
    v = (kp < NBINS) ? (w * cosf(th) * (1.0f / 512.0f))
                     : (-w * sinf(th) * (1.0f / 512.0f));
  }
  Bi[idx] = v;
}

// ---------------------------------------------------------------------------
// f32 WMMA GEMM: C[M x N] = A[M x K] * B[K x N].
// One wave computes a 16x64 tile of C (4 accumulators) so the A fragment is
// reused across 4 V_WMMA_F32_16X16X4_F32 issues: 9 loads / 4 WMMAs.
// Fragment layouts per CDNA5 ISA 7.12.2:
//   A 16x4 : lanes 0-15 rows, VGPR{0,1} = K{0,1}; lanes 16-31 = K{2,3}
//   B 4x16 : lanes 0-15 cols, VGPR{0,1} = K{0,1}; lanes 16-31 = K{2,3}
//   C      : VGPR j -> row j (lanes 0-15) / row 8+j (lanes 16-31)
// ---------------------------------------------------------------------------
__global__ void ks_gemm16x64(const float* __restrict__ A, const float* __restrict__ B,
                             float* __restrict__ C, int tilesM, int groupsN,
                             int K, int lda, int ldb, int ldc) {
  int wave = blockIdx.x * (blockDim.x >> 5) + (threadIdx.x >> 5);
  if (wave >= tilesM * groupsN) return;
  int lane = threadIdx.x & 31;
  int tm = wave / groupsN;
  int tg = wave % groupsN;
  int r  = lane & 15;
  int hi = lane >> 4;

  const float* Arow = A + (size_t)(tm * 16 + r) * lda + 2 * hi;
  const float* B0   = B + (size_t)(2 * hi) * ldb + tg * 64 + r;

  v8f acc0 = {0.f,0.f,0.f,0.f,0.f,0.f,0.f,0.f};
  v8f acc1 = acc0, acc2 = acc0, acc3 = acc0;

  for (int k = 0; k < K; k += 4) {
    v2f a;
    a.x = Arow[k];
    a.y = Arow[k + 1];
    const float* Bk = B0 + (size_t)k * ldb;
    v2f b0, b1, b2, b3;
    b0.x = Bk[0];   b0.y = Bk[(size_t)ldb + 0];
    b1.x = Bk[16];  b1.y = Bk[(size_t)ldb + 16];
    b2.x = Bk[32];  b2.y = Bk[(size_t)ldb + 32];
    b3.x = Bk[48];  b3.y = Bk[(size_t)ldb + 48];
    acc0 = __builtin_amdgcn_wmma_f32_16x16x4_f32(false, a, false, b0, (short)0, acc0, false, false);
    acc1 = __builtin_amdgcn_wmma_f32_16x16x4_f32(false, a, false, b1, (short)0, acc1, false, false);
    acc2 = __builtin_amdgcn_wmma_f32_16x16x4_f32(false, a, false, b2, (short)0, acc2, false, false);
    acc3 = __builtin_amdgcn_wmma_f32_16x16x4_f32(false, a, false, b3, (short)0, acc3, false, false);
  }
  float* Crow = C + (size_t)(tm * 16 + hi * 8) * ldc + tg * 64 + r;
#pragma unroll
  for (int j = 0; j < 8; ++j) {
    Crow[(size_t)j * ldc +  0] = acc0[j];
    Crow[(size_t)j * ldc + 16] = acc1[j];
    Crow[(size_t)j * ldc + 32] = acc2[j];
    Crow[(size_t)j * ldc + 48] = acc3[j];
  }
}

// ---------------------------------------------------------------------------
// Single-WAVE frequency-domain serial scan (wave32).  Lane l owns bins
// [9l, 9l+9) in registers; lanes 29-31 idle (clamped loads, guarded stores).
// Periodic Hamming => 3-tap stencil in frequency:
//   U = S_t * tf_t ;  S_{t+1}[k] = 0.54*U[k] - 0.23*(U[k-1]+U[k+1]) + P[t+1][k]
// Cross-lane halo = 4 shuffles/step; conj-symmetric edges are lane-local:
//   U[-1] = conj(U[1])  (lane 0),  U[257] = conj(U[255])  (lane 28, j=4).
// No barriers, no LDS; tf/P rows prefetched ~8 steps ahead.
// ---------------------------------------------------------------------------
__global__ void __launch_bounds__(32)
ks_scan(const float* __restrict__ tfr, const float* __restrict__ tfi,
        const float* __restrict__ P, float* __restrict__ S) {
  int lane = threadIdx.x;
  int base = lane * 9;
  float sr[9], si[9];
#pragma unroll
  for (int j = 0; j < 9; ++j) {                 // S_0 = P_0
    int k  = base + j;
    int kc = min(k, NBINS - 1);
    sr[j] = P[kc];
    si[j] = P[NBINS + kc];
    if (k < NBINS) { S[k] = sr[j]; S[NBINS + k] = si[j]; }
  }
  for (int t = 0; t < N_STEPS; ++t) {
    if (t + 8 < N_STEPS) {
      __builtin_prefetch(tfr + (size_t)(t + 8) * NBINS + base, 0, 1);
      __builtin_prefetch(tfi + (size_t)(t + 8) * NBINS + base, 0, 1);
      __builtin_prefetch(P + (size_t)(t + 8) * PCOLS + base, 0, 1);
    }
    const float* trRow = tfr + (size_t)t * NBINS;
    const float* tiRow = tfi + (size_t)t * NBINS;
    float ur[9], ui[9];
#pragma unroll
    for (int j = 0; j < 9; ++j) {
      int kc = min(base + j, NBINS - 1);
      float tr = trRow[kc];
      float ti = tiRow[kc];
      ur[j] = sr[j] * tr - si[j] * ti;
      ui[j] = sr[j] * ti + si[j] * tr;
    }
    // halo exchange: left neighbor's last bin, right neighbor's first bin
    float plr = __shfl_up(ur[8], 1, 32);
    float pli = __shfl_up(ui[8], 1, 32);
    float nrr = __shfl_down(ur[0], 1, 32);
    float nri = __shfl_down(ui[0], 1, 32);
    if (lane == 0) { plr = ur[1]; pli = -ui[1]; }   // bin -1 = conj(bin 1)
    bool haveP = (t + 1 < N_STEPS);                 // last step adds no impulse
    const float* Prow = P + (size_t)(t + 1) * PCOLS;
    float* Srow = S + (size_t)(t + 1) * PCOLS;
#pragma unroll
    for (int j = 0; j < 9; ++j) {
      int k = base + j;
      float lr = (j == 0) ? plr : ur[j - 1];
      float li = (j == 0) ? pli : ui[j - 1];
      float rr = (j == 8) ? nrr : ur[j + 1];
      float ri = (j == 8) ? nri : ui[j + 1];
      if (k == 256) { rr = ur[3]; ri = -ui[3]; }    // bin 257 = conj(bin 255)
      float pr = 0.0f, pi = 0.0f;
      int kc = min(k, NBINS - 1);
      if (haveP) { pr = Prow[kc]; pi = Prow[NBINS + kc]; }
      float nsr = 0.54f * ur[j] - 0.23f * (lr + rr) + pr;
      float nsi = 0.54f * ui[j] - 0.23f * (li + ri) + pi;
      sr[j] = nsr;
      si[j] = nsi;
      if (k < NBINS) { Srow[k] = nsr; Srow[NBINS + k] = nsi; }
    }
  }
}

// ---------------------------------------------------------------------------
// Overlap-add: each output sample is the sum of exactly 2 block taps (hop 256)
// ---------------------------------------------------------------------------
__global__ void ks_ola(const float* __restrict__ blocks, float* __restrict__ out) {
  int n = blockIdx.x * blockDim.x + threadIdx.x;
  if (n >= PIANO_SAMPLES) return;
  int t1  = n >> 8;
  int off = n & 255;
  float v = blocks[(size_t)t1 * BLOCK_N + off];
  if (t1 >= 1) v += blocks[(size_t)(t1 - 1) * BLOCK_N + off + 256];
  out[n] = v;
}

// ---------------------------------------------------------------------------
extern "C" void kernel_launch(void* const* d_in, const int* in_sizes, int n_in,
                              void* d_out, int out_size, void* d_ws, size_t ws_size,
                              hipStream_t stream) {
  (void)in_sizes; (void)n_in; (void)out_size; (void)ws_size;
  const float* ev    = (const float*)d_in[1];
  const float* tfr   = (const float*)d_in[2];
  const float* tfi   = (const float*)d_in[3];
  const float* er    = (const float*)d_in[4];
  const float* ei    = (const float*)d_in[5];
  const float* noise = (const float*)d_in[6];
  float* out = (float*)d_out;

  float* ws = (float*)d_ws;
  float* Bf = ws;                               // 512   x 576
  float* Bi = Bf + (size_t)512 * PCOLS;         // 516   x 512
  float* X  = Bi + (size_t)KPAD * 512;          // 10240 x 512
  float* P  = X  + (size_t)N_STEPS * BLOCK_N;   // 10240 x 576
  float* S  = P  + (size_t)N_STEPS * PCOLS;     // 10256 x 576 (zero-padded)
  float* BL = S  + (size_t)SROWS * PCOLS;       // 10256 x 512 time-domain blocks

  // 1) excitation windows * noise
  {
    int n = N_STEPS * BLOCK_N;
    ks_gen_x<<<(n + 255) / 256, 256, 0, stream>>>(ev, noise, X);
  }
  // 2) bases
  {
    int n = 512 * PCOLS;
    ks_gen_bf<<<(n + 255) / 256, 256, 0, stream>>>(er, ei, Bf);
    n = KPAD * 512;
    ks_gen_bi<<<(n + 255) / 256, 256, 0, stream>>>(Bi);
  }
  // 3) zero S so K/N padding and pad rows read exact zeros
  {
    int n = SROWS * PCOLS;
    ks_zero<<<(n + 255) / 256, 256, 0, stream>>>(S, n);
  }
  // 4) Phase A GEMM: P[10240 x 576] = X[10240 x 512] * Bf[512 x 576]   (WMMA f32)
  {
    int tilesM = N_STEPS / 16, groupsN = PCOLS / 64;         // 640 x 9
    int waves = tilesM * groupsN;
    ks_gemm16x64<<<(waves + 7) / 8, 256, 0, stream>>>(X, Bf, P, tilesM, groupsN,
                                                      512, 512, PCOLS, PCOLS);
  }
  // 5) serial frequency-domain scan (single wave32, barrier-free)
  ks_scan<<<1, 32, 0, stream>>>(tfr, tfi, P, S);
  // 6) Phase C GEMM: BL[10256 x 512] = S[10256 x 516] * Bi[516 x 512]  (WMMA f32)
  {
    int tilesM = SROWS / 16, groupsN = 512 / 64;             // 641 x 8
    int waves = tilesM * groupsN;
    ks_gemm16x64<<<(waves + 7) / 8, 256, 0, stream>>>(S, Bi, BL, tilesM, groupsN,
                                                      KPAD, PCOLS, 512, 512);
  }
  // 7) overlap-add
  ks_ola<<<(PIANO_SAMPLES + 255) / 256, 256, 0, stream>>>(BL, out);
}